// GNN_111669150109
// MI455X (gfx1250) — compile-verified
//
#include <hip/hip_runtime.h>
#include <hip/hip_bf16.h>
#include <math.h>

// ---------------------------------------------------------------------------
// GNN (GCN x4) for MI455X / gfx1250, wave32, WMMA f32 16x16x4 (full precision)
// GEMM: 64-row macro-tile per block, B operand register-resident across tiles,
// A tile staged to LDS via GLOBAL_LOAD_ASYNC_TO_LDS_B128 (ASYNCcnt-tracked).
// ---------------------------------------------------------------------------

#define HDIM 128           // F == H == 128
#define NNODES 50000
#define NEDGES 800000
#define NGRAPH 512
#define DEPTH 4
#define BN_EPS 1e-5f

typedef __attribute__((ext_vector_type(2))) float v2f;
typedef __attribute__((ext_vector_type(8))) float v8f;
typedef __attribute__((ext_vector_type(4))) int   v4i;

#define USE_ASYNC_LDS 0
#if defined(__has_builtin)
#if __has_builtin(__builtin_amdgcn_global_load_async_to_lds_b128) && \
    __has_builtin(__builtin_amdgcn_s_wait_asynccnt)
#undef USE_ASYNC_LDS
#define USE_ASYNC_LDS 1
// Builtin signature (from clang diagnostic): (v4i addrspace(1)*, v4i addrspace(3)*, imm, imm)
typedef __attribute__((address_space(1))) v4i v4i_g;
typedef __attribute__((address_space(3))) v4i v4i_l;
#endif
#endif

// -------------------------- small utility kernels --------------------------

__global__ void zero_f32_kernel(float* __restrict__ p, int n) {
    int i = blockIdx.x * blockDim.x + threadIdx.x;
    if (i < n) p[i] = 0.0f;
}

__global__ void degree_kernel(const int* __restrict__ dst, float* __restrict__ deg, int nEdges) {
    int e = blockIdx.x * blockDim.x + threadIdx.x;
    if (e < nEdges) unsafeAtomicAdd(&deg[dst[e]], 1.0f);
}

__global__ void dinv_kernel(float* __restrict__ deg_to_dinv, int nNodes) {
    int n = blockIdx.x * blockDim.x + threadIdx.x;
    if (n < nNodes) deg_to_dinv[n] = rsqrtf(deg_to_dinv[n] + 1.0f);  // +1 self loop
}

// ------------------------------ WMMA GEMM ----------------------------------
// Out[r, c] = sum_k A[r, k] * W[k, c] (+ bias[c]); A: [nRows x 128] row-major,
// W: [128 x 128] row-major. Block = 256 thr = 8 waves; block owns a 64-row
// macro tile; wave w owns columns [16w, 16w+16) and iterates 4 row tiles with
// its full 128x16 W slice held in registers. 128 v_wmma_f32_16x16x4 per wave.
__global__ __launch_bounds__(256)
void gemm_h128_wmma_kernel(const float* __restrict__ A, const float* __restrict__ W,
                           const float* __restrict__ bias, float* __restrict__ Out,
                           int nRows) {
    __shared__ float As[64][HDIM + 4];   // +4 pad: stride 132 -> bank-conflict free
    const int tid = threadIdx.x;
    const int rowBase = blockIdx.x * 64;

    // Cooperative 64x128 A tile load: 2048 float4 / 256 threads = 8 each.
    for (int i = tid; i < 64 * (HDIM / 4); i += 256) {
        const int r  = i >> 5;            // / (HDIM/4)
        const int c4 = (i & 31) * 4;      // % (HDIM/4) * 4
        const int gr = rowBase + r;
#if USE_ASYNC_LDS
        if (gr < nRows) {
            // ASYNCcnt-tracked DMA: global -> LDS, no VGPR round trip.
            __builtin_amdgcn_global_load_async_to_lds_b128(
                (v4i_g*)(A + (size_t)gr * HDIM + c4),
                (v4i_l*)&As[r][c4], 0, 0);
        } else {
            As[r][c4 + 0] = 0.f; As[r][c4 + 1] = 0.f;
            As[r][c4 + 2] = 0.f; As[r][c4 + 3] = 0.f;
        }
#else
        float4 v = make_float4(0.f, 0.f, 0.f, 0.f);
        if (gr < nRows) v = *(const float4*)(A + (size_t)gr * HDIM + c4);
        As[r][c4 + 0] = v.x; As[r][c4 + 1] = v.y;
        As[r][c4 + 2] = v.z; As[r][c4 + 3] = v.w;
#endif
    }
#if USE_ASYNC_LDS
    __builtin_amdgcn_s_wait_asynccnt(0);
#endif
    __syncthreads();

    const int wave = tid >> 5;
    const int lane = tid & 31;
    const int colBase = wave * 16;
    const int m  = lane & 15;            // A-row within tile / B,C,D-col within tile
    const int kk = (lane >> 4) << 1;     // K sub-offset: lanes 0-15 -> {0,1}, 16-31 -> {2,3}

    // B operand (this wave's 128x16 slice of W): 64 VGPRs, loaded once,
    // reused across all 4 row tiles.
    const float* wp = W + colBase + m;   // column n = colBase + m
    __builtin_prefetch(wp, 0, 3);        // global_prefetch_b8 into near caches
    v2f breg[32];
#pragma unroll
    for (int k0 = 0; k0 < HDIM; k0 += 4) {
        breg[k0 >> 2][0] = wp[(size_t)(k0 + kk)     * HDIM];
        breg[k0 >> 2][1] = wp[(size_t)(k0 + kk + 1) * HDIM];
    }

    const int col  = colBase + m;
    const int rOff = (lane >> 4) * 8;
    const float bv = bias ? bias[col] : 0.0f;

#pragma unroll
    for (int t = 0; t < 4; ++t) {
        v8f c = {};
#pragma unroll
        for (int k0 = 0; k0 < HDIM; k0 += 4) {
            v2f a;
            const float2 av = *(const float2*)&As[t * 16 + m][k0 + kk]; // 8B aligned
            a[0] = av.x;
            a[1] = av.y;
            // D = A x B + C, fp32 in / fp32 accumulate
            c = __builtin_amdgcn_wmma_f32_16x16x4_f32(false, a, false, breg[k0 >> 2],
                                                      (short)0, c, false, false);
        }
        // C/D layout: VGPR i -> row (i + 8*(lane>>4)), col = lane&15
        const int rowT = rowBase + t * 16 + rOff;
#pragma unroll
        for (int i = 0; i < 8; ++i) {
            const int r = rowT + i;
            if (r < nRows) Out[(size_t)r * HDIM + col] = c[i] + bv;
        }
    }
}

// --------------------------- GCN aggregation -------------------------------

// agg[n,:] = dinv[n]^2 * hw[n,:] + bias  (self-loop term; initializes agg)
__global__ void agg_init_kernel(const float* __restrict__ hw, const float* __restrict__ dinv,
                                const float* __restrict__ bias, float* __restrict__ agg,
                                int nNodes) {
    const int gid = blockIdx.x * blockDim.x + threadIdx.x;   // over nNodes * 32 chunks
    if (gid >= nNodes * (HDIM / 4)) return;
    const int n  = gid / (HDIM / 4);
    const int f  = (gid % (HDIM / 4)) * 4;
    const float d = dinv[n];
    const float s = d * d;
    const float4 v = *(const float4*)(hw + (size_t)n * HDIM + f);
    const float4 b = *(const float4*)(bias + f);
    float4 o;
    o.x = fmaf(s, v.x, b.x); o.y = fmaf(s, v.y, b.y);
    o.z = fmaf(s, v.z, b.z); o.w = fmaf(s, v.w, b.w);
    *(float4*)(agg + (size_t)n * HDIM + f) = o;
}

// agg[dst,:] += dinv[src]*dinv[dst] * hw[src,:]  (one thread per edge x float4)
__global__ void edge_scatter_kernel(const int* __restrict__ src, const int* __restrict__ dst,
                                    const float* __restrict__ dinv, const float* __restrict__ hw,
                                    float* __restrict__ agg, int nEdges) {
    const int gid = blockIdx.x * blockDim.x + threadIdx.x;   // over nEdges * 32
    if (gid >= nEdges * 32) return;
    const int e = gid >> 5;
    const int f = (gid & 31) * 4;
    const int s = src[e];
    const int d = dst[e];
    const float nrm = dinv[s] * dinv[d];
    const float4 v = *(const float4*)(hw + (size_t)s * HDIM + f);
    float* ap = agg + (size_t)d * HDIM + f;
    unsafeAtomicAdd(ap + 0, nrm * v.x);
    unsafeAtomicAdd(ap + 1, nrm * v.y);
    unsafeAtomicAdd(ap + 2, nrm * v.z);
    unsafeAtomicAdd(ap + 3, nrm * v.w);
}

// ------------------------------ ReLU + BN ----------------------------------

// In-place ReLU + per-feature {sum, sumsq} partials. 128 thr/block, thread = feature.
__global__ __launch_bounds__(128)
void bn_stats_kernel(float* __restrict__ h, float* __restrict__ stats, int nNodes) {
    const int f = threadIdx.x;
    float s1 = 0.0f, s2 = 0.0f;
    for (int r = blockIdx.x; r < nNodes; r += gridDim.x) {
        float v = h[(size_t)r * HDIM + f];
        v = fmaxf(v, 0.0f);
        h[(size_t)r * HDIM + f] = v;
        s1 += v;
        s2 += v * v;
    }
    unsafeAtomicAdd(&stats[f],        s1);
    unsafeAtomicAdd(&stats[HDIM + f], s2);
}

__global__ __launch_bounds__(128)
void bn_coef_kernel(const float* __restrict__ stats, const float* __restrict__ gamma,
                    const float* __restrict__ beta, float* __restrict__ coef, float invN) {
    const int f = threadIdx.x;
    const float mu  = stats[f] * invN;
    const float var = stats[HDIM + f] * invN - mu * mu;
    const float sc  = rsqrtf(var + BN_EPS) * gamma[f];
    coef[f]        = sc;
    coef[HDIM + f] = beta[f] - mu * sc;
}

__global__ void bn_apply_kernel(float* __restrict__ h, const float* __restrict__ coef,
                                int nNodes) {
    const int gid = blockIdx.x * blockDim.x + threadIdx.x;   // over nNodes * 32
    if (gid >= nNodes * (HDIM / 4)) return;
    const int n = gid / (HDIM / 4);
    const int f = (gid % (HDIM / 4)) * 4;
    float4 v = *(float4*)(h + (size_t)n * HDIM + f);
    const float4 sc = *(const float4*)(coef + f);
    const float4 sh = *(const float4*)(coef + HDIM + f);
    v.x = fmaf(v.x, sc.x, sh.x); v.y = fmaf(v.y, sc.y, sh.y);
    v.z = fmaf(v.z, sc.z, sh.z); v.w = fmaf(v.w, sc.w, sh.w);
    *(float4*)(h + (size_t)n * HDIM + f) = v;
}

// ------------------------------- pooling -----------------------------------

__global__ void pool_count_kernel(const int* __restrict__ batch, float* __restrict__ counts,
                                  int nNodes) {
    const int n = blockIdx.x * blockDim.x + threadIdx.x;
    if (n < nNodes) unsafeAtomicAdd(&counts[batch[n]], 1.0f);
}

__global__ void pool_sum_kernel(const float* __restrict__ h, const int* __restrict__ batch,
                                float* __restrict__ pooled, int nNodes) {
    const int gid = blockIdx.x * blockDim.x + threadIdx.x;   // over nNodes * 32
    if (gid >= nNodes * (HDIM / 4)) return;
    const int n = gid / (HDIM / 4);
    const int f = (gid % (HDIM / 4)) * 4;
    const int g = batch[n];
    const float4 v = *(const float4*)(h + (size_t)n * HDIM + f);
    float* pp = pooled + (size_t)g * HDIM + f;
    unsafeAtomicAdd(pp + 0, v.x);
    unsafeAtomicAdd(pp + 1, v.y);
    unsafeAtomicAdd(pp + 2, v.z);
    unsafeAtomicAdd(pp + 3, v.w);
}

__global__ void pool_div_kernel(const float* __restrict__ pooled, const float* __restrict__ counts,
                                float* __restrict__ out, int total) {
    const int i = blockIdx.x * blockDim.x + threadIdx.x;
    if (i < total) out[i] = pooled[i] / fmaxf(counts[i >> 7], 1.0f);
}

// ------------------------------ host driver --------------------------------

extern "C" void kernel_launch(void* const* d_in, const int* in_sizes, int n_in,
                              void* d_out, int out_size, void* d_ws, size_t ws_size,
                              hipStream_t stream) {
    const float* x        = (const float*)d_in[0];
    const int*   eidx     = (const int*)d_in[1];   // [2, E]
    const int*   batch    = (const int*)d_in[2];   // [N]
    const float* W_emb    = (const float*)d_in[3];
    const float* b_emb    = (const float*)d_in[4];
    const float* W_conv   = (const float*)d_in[5]; // [DEPTH, H, H]
    const float* b_conv   = (const float*)d_in[6]; // [DEPTH, H]
    const float* gamma    = (const float*)d_in[7]; // [DEPTH-1, H]
    const float* beta     = (const float*)d_in[8]; // [DEPTH-1, H]
    float* out = (float*)d_out;

    const int* src = eidx;
    const int* dst = eidx + NEDGES;

    // Workspace carve-up (floats)
    const size_t NH = (size_t)NNODES * HDIM;       // 6.4M floats
    float* ws     = (float*)d_ws;
    float* hA     = ws;                            // [N, H]
    float* hB     = hA + NH;                       // [N, H]
    float* hw     = hB + NH;                       // [N, H]
    float* dinv   = hw + NH;                       // [N]  (deg -> dinv in place)
    float* stats  = dinv + 50048;                  // [256] {sum, sumsq}
    float* coef   = stats + 256;                   // [256] {scale, shift}
    float* pooled = coef + 256;                    // [G, H]
    float* counts = pooled + (size_t)NGRAPH * HDIM;// [G]

    const int TPB = 256;
    const dim3 blk(TPB);
    const int gemmGrid  = (NNODES + 63) / 64;                  // 782
    const int nodeVec   = (NNODES * (HDIM / 4) + TPB - 1) / TPB;
    const int edgeVec   = (NEDGES * 32 + TPB - 1) / TPB;

    // ---- degrees & symmetric normalization ----
    zero_f32_kernel<<<(NNODES + TPB - 1) / TPB, blk, 0, stream>>>(dinv, NNODES);
    degree_kernel<<<(NEDGES + TPB - 1) / TPB, blk, 0, stream>>>(dst, dinv, NEDGES);
    dinv_kernel<<<(NNODES + TPB - 1) / TPB, blk, 0, stream>>>(dinv, NNODES);

    // ---- embedding: h = x @ W_emb + b_emb ----
    gemm_h128_wmma_kernel<<<gemmGrid, blk, 0, stream>>>(x, W_emb, b_emb, hA, NNODES);

    // ---- GCN layers ----
    float* hCur = hA;
    float* hNxt = hB;
    for (int i = 0; i < DEPTH; ++i) {
        gemm_h128_wmma_kernel<<<gemmGrid, blk, 0, stream>>>(
            hCur, W_conv + (size_t)i * HDIM * HDIM, nullptr, hw, NNODES);
        agg_init_kernel<<<nodeVec, blk, 0, stream>>>(
            hw, dinv, b_conv + (size_t)i * HDIM, hNxt, NNODES);
        edge_scatter_kernel<<<edgeVec, blk, 0, stream>>>(
            src, dst, dinv, hw, hNxt, NEDGES);
        if (i < DEPTH - 1) {
            zero_f32_kernel<<<1, blk, 0, stream>>>(stats, 256);
            bn_stats_kernel<<<512, dim3(128), 0, stream>>>(hNxt, stats, NNODES);
            bn_coef_kernel<<<1, dim3(128), 0, stream>>>(
                stats, gamma + (size_t)i * HDIM, beta + (size_t)i * HDIM,
                coef, 1.0f / (float)NNODES);
            bn_apply_kernel<<<nodeVec, blk, 0, stream>>>(hNxt, coef, NNODES);
        }
        float* t = hCur; hCur = hNxt; hNxt = t;
    }

    // ---- global mean pool ----
    const int poolN = NGRAPH * HDIM + NGRAPH;      // pooled + counts contiguous
    zero_f32_kernel<<<(poolN + TPB - 1) / TPB, blk, 0, stream>>>(pooled, poolN);
    pool_count_kernel<<<(NNODES + TPB - 1) / TPB, blk, 0, stream>>>(batch, counts, NNODES);
    pool_sum_kernel<<<nodeVec, blk, 0, stream>>>(hCur, batch, pooled, NNODES);
    pool_div_kernel<<<(NGRAPH * HDIM + TPB - 1) / TPB, blk, 0, stream>>>(
        pooled, counts, out, NGRAPH * HDIM);

    (void)in_sizes; (void)n_in; (void)out_size; (void)ws_size;
}